// TaskAlignedAssigner_63977832842029
// MI455X (gfx1250) — compile-verified
//
#include <hip/hip_runtime.h>
#include <stdint.h>

#define BS    16
#define NA    8400
#define NGT   64
#define NC    80
#define TOPK  13
#define ACHUNK 525            // anchors per TDM tile (525*16 = 8400)
#define NCH    16
#define CELEMS (ACHUNK * 4)   // floats per tile
#define ABLK   256
#define CAND   (ABLK * TOPK)

typedef unsigned int u32;
typedef unsigned long long u64;
typedef u32 v4u __attribute__((ext_vector_type(4)));
typedef int  v8i __attribute__((ext_vector_type(8)));
typedef int  v4i __attribute__((ext_vector_type(4)));
typedef float v2f __attribute__((ext_vector_type(2)));
typedef float v8f __attribute__((ext_vector_type(8)));

// ---------------------------------------------------------------- CIoU (reference formula, box1 = gt, box2 = pred)
__device__ __forceinline__ float ciou_gt_pd(float gx1, float gy1, float gx2, float gy2,
                                            float px1, float py1, float px2, float py2) {
  const float eps = 1e-7f;
  float w1 = gx2 - gx1, h1 = gy2 - gy1 + eps;
  float w2 = px2 - px1, h2 = py2 - py1 + eps;
  float iw = fminf(gx2, px2) - fmaxf(gx1, px1);
  float ih = fminf(gy2, py2) - fmaxf(gy1, py1);
  float inter = fmaxf(iw, 0.0f) * fmaxf(ih, 0.0f);
  float uni = w1 * h1 + w2 * h2 - inter + eps;
  float iou = inter / uni;
  float cw = fmaxf(gx2, px2) - fminf(gx1, px1);
  float ch = fmaxf(gy2, py2) - fminf(gy1, py1);
  float c2 = cw * cw + ch * ch + eps;
  float dx = px1 + px2 - gx1 - gx2;
  float dy = py1 + py2 - gy1 - gy2;
  float rho2 = (dx * dx + dy * dy) * 0.25f;
  float dat = atanf(w2 / h2) - atanf(w1 / h1);
  float v = 0.405284734569351f * dat * dat;   // 4/pi^2
  float al = v / (v - iou + (1.0f + eps));
  return iou - (rho2 / c2 + v * al);
}

__device__ __forceinline__ float min4f(float a, float b, float c, float d) {
  return fminf(fminf(a, b), fminf(c, d));
}

// ---------------------------------------------------------------- TDM: async DMA of nelem f32 from global -> LDS
// D# group0: count=1, lds_addr, 57-bit global addr, type=2.  group1: data_size=4B,
// tensor_dim0 = tile_dim0 = nelem (1 row, no OOB), tensor_dim1 = tile_dim1 = 1.
__device__ __forceinline__ void tdm_load_f32(const float* gsrc, u32 lds_byte, int nelem) {
  u64 ga = (u64)(uintptr_t)gsrc;
  v4u g0;
  g0.x = 1u;                                             // count = 1 valid descriptor
  g0.y = lds_byte;                                       // LDS byte address
  g0.z = (u32)ga;                                        // global_addr[31:0]
  g0.w = ((u32)(ga >> 32) & 0x01FFFFFFu) | 0x80000000u;  // global_addr[56:32] | type=2
  v8i g1;
  g1.s0 = 0x00020000;                                    // wg_mask=0, data_size=2 (4B)
  g1.s1 = (int)((u32)nelem << 16);                       // tensor_dim0[15:0]
  g1.s2 = (int)(0x00010000u | ((u32)nelem >> 16));       // tensor_dim0[31:16], tensor_dim1=1
  g1.s3 = (int)((u32)nelem << 16);                       // tensor_dim1[31:16]=0, tile_dim0=nelem
  g1.s4 = 1;                                             // tile_dim1=1, tile_dim2=0
  g1.s5 = nelem;                                         // tensor_dim0_stride[31:0]
  g1.s6 = 0;
  g1.s7 = 0;
  v4i z4 = (v4i)0;                                       // groups 2/3 unused (2D tensor)
  v8i z8 = (v8i)0;
  __builtin_amdgcn_tensor_load_to_lds(g0, g1, z4, z4, z8, 0);
}

// ---------------------------------------------------------------- workspace zero
__global__ void initWS(u32* p, int nwords) {
  int i = blockIdx.x * blockDim.x + threadIdx.x;
  if (i < nwords) p[i] = 0u;
}

// ---------------------------------------------------------------- Phase A: per (b, gt): top-13 selection
__global__ __launch_bounds__(ABLK) void phaseA(const float* __restrict__ pd_scores,
                                               const float* __restrict__ pd_bboxes,
                                               const float* __restrict__ anc,
                                               const int*   __restrict__ gt_labels,
                                               const float* __restrict__ gt_bboxes,
                                               const float* __restrict__ mask_gt,
                                               int* __restrict__ selcount,
                                               u64* __restrict__ selbits) {
  __shared__ __align__(16) float sbox[2][CELEMS];   // double-buffered pred-box tiles
  __shared__ float cval[CAND];
  __shared__ int   cidx[CAND];
  __shared__ float rv[ABLK];
  __shared__ int   ri[ABLK];
  __shared__ int   rj[ABLK];
  __shared__ int   selA[TOPK];

  const int tid = threadIdx.x;
  const int row = blockIdx.x;          // row = b*64 + n
  const int b = row >> 6;
  const int n = row & 63;

  if (mask_gt[row] <= 0.0f) return;    // row contributes nothing (topk dedup kills it)

  const float4 g = ((const float4*)gt_bboxes)[row];
  const float gx1 = g.x, gy1 = g.y, gx2 = g.z, gy2 = g.w;
  const int lbl = gt_labels[row];
  const float* srcb = pd_scores + (size_t)b * NA * NC;
  const float* boxb = pd_bboxes + (size_t)b * NA * 4;
  const float2* anc2 = (const float2*)anc;

  // per-thread sorted top-13 (value desc, index asc)
  float tv[TOPK];
  int   ti[TOPK];
#pragma unroll
  for (int k = 0; k < TOPK; ++k) { tv[k] = -1.0f; ti[k] = 0x7fffffff; }

  if (tid < 32) tdm_load_f32(boxb, (u32)(u64)(uintptr_t)(const void*)&sbox[0][0], CELEMS);

  for (int c = 0; c < NCH; ++c) {
    if (tid < 32) {
      if (c + 1 < NCH) {
        tdm_load_f32(boxb + (size_t)(c + 1) * CELEMS,
                     (u32)(u64)(uintptr_t)(const void*)&sbox[(c + 1) & 1][0], CELEMS);
        __builtin_amdgcn_s_wait_tensorcnt(1);   // tile c done (in-order TDM)
      } else {
        __builtin_amdgcn_s_wait_tensorcnt(0);
      }
    }
    __syncthreads();
    const float* buf = sbox[c & 1];
    for (int l = tid; l < ACHUNK; l += ABLK) {
      int a = c * ACHUNK + l;
      float2 ap = anc2[a];
      float mn = min4f(ap.x - gx1, ap.y - gy1, gx2 - ap.x, gy2 - ap.y);
      float al = 0.0f;
      if (mn > 1e-9f) {
        float4 p = ((const float4*)buf)[l];
        float ov = fmaxf(ciou_gt_pd(gx1, gy1, gx2, gy2, p.x, p.y, p.z, p.w), 0.0f);
        float sc = srcb[(size_t)a * NC + lbl];
        float o2 = ov * ov;
        al = sc * o2 * o2 * o2;
      }
      if (al > tv[TOPK - 1] || (al == tv[TOPK - 1] && a < ti[TOPK - 1])) {
        tv[TOPK - 1] = al; ti[TOPK - 1] = a;
#pragma unroll
        for (int k = TOPK - 1; k > 0; --k) {
          bool sw = tv[k] > tv[k - 1] || (tv[k] == tv[k - 1] && ti[k] < ti[k - 1]);
          if (sw) {
            float fv = tv[k]; tv[k] = tv[k - 1]; tv[k - 1] = fv;
            int iv = ti[k]; ti[k] = ti[k - 1]; ti[k - 1] = iv;
          }
        }
      }
    }
    __syncthreads();
  }

  // dump candidates, then 13 x block-argmax merge (exact lax.top_k tie-breaking)
#pragma unroll
  for (int k = 0; k < TOPK; ++k) { cval[tid + ABLK * k] = tv[k]; cidx[tid + ABLK * k] = ti[k]; }
  __syncthreads();

  for (int s = 0; s < TOPK; ++s) {
    float bv = -2.0f; int bi = 0x7fffffff; int bj = 0;
    for (int j = tid; j < CAND; j += ABLK) {
      float v = cval[j]; int id = cidx[j];
      if (v > bv || (v == bv && id < bi)) { bv = v; bi = id; bj = j; }
    }
    rv[tid] = bv; ri[tid] = bi; rj[tid] = bj;
    __syncthreads();
    for (int st = ABLK / 2; st > 0; st >>= 1) {
      if (tid < st) {
        float v2 = rv[tid + st]; int i2 = ri[tid + st];
        if (v2 > rv[tid] || (v2 == rv[tid] && i2 < ri[tid])) {
          rv[tid] = v2; ri[tid] = i2; rj[tid] = rj[tid + st];
        }
      }
      __syncthreads();
    }
    if (tid == 0) { selA[s] = ri[0]; cval[rj[0]] = -2.0f; }
    __syncthreads();
  }

  // mask_pos = topk-membership * in_gts * mask_gt  ->  sparse scatter
  if (tid < TOPK) {
    int a = selA[tid];
    float2 ap = anc2[a];
    float mn = min4f(ap.x - gx1, ap.y - gy1, gx2 - ap.x, gy2 - ap.y);
    if (mn > 1e-9f) {
      atomicAdd(&selcount[(size_t)b * NA + a], 1);
      atomicOr(&selbits[(size_t)b * NA + a], 1ull << n);
    }
  }
}

// ---------------------------------------------------------------- Phase B1: per-anchor conflict resolution + targets
__global__ __launch_bounds__(256) void phaseB1(const float* __restrict__ pd_scores,
                                               const float* __restrict__ pd_bboxes,
                                               const float* __restrict__ anc,
                                               const int*   __restrict__ gt_labels,
                                               const float* __restrict__ gt_bboxes,
                                               const float* __restrict__ mask_gt,
                                               const int*   __restrict__ selcount,
                                               const u64*   __restrict__ selbits,
                                               float* __restrict__ pos_align,
                                               float* __restrict__ pos_ov,
                                               float* __restrict__ ws_align,
                                               int*   __restrict__ ws_tgi,
                                               int*   __restrict__ ws_fg,
                                               float* __restrict__ outL,
                                               float* __restrict__ outB,
                                               float* __restrict__ outF,
                                               float* __restrict__ outG) {
  int i = blockIdx.x * 256 + threadIdx.x;
  if (i >= BS * NA) return;
  int b = i / NA, a = i - b * NA;
  int cnt = selcount[i];
  int fg = (cnt > 0) ? 1 : 0;
  int tgi = 0;
  float ov = 0.0f;

  float4 pb; float2 ap;
  if (cnt > 0) {
    pb = ((const float4*)pd_bboxes)[i];
    ap = ((const float2*)anc)[a];
  }

  if (cnt > 1) {
    // is_max: argmax over all gts of masked overlaps (first-max tie like jnp.argmax)
    float best = -1.0f; int bn = 0;
    for (int nn = 0; nn < NGT; ++nn) {
      int r = b * NGT + nn;
      float o = 0.0f;
      if (mask_gt[r] > 0.0f) {
        float4 g = ((const float4*)gt_bboxes)[r];
        float mn = min4f(ap.x - g.x, ap.y - g.y, g.z - ap.x, g.w - ap.y);
        if (mn > 1e-9f)
          o = fmaxf(ciou_gt_pd(g.x, g.y, g.z, g.w, pb.x, pb.y, pb.z, pb.w), 0.0f);
      }
      if (o > best) { best = o; bn = nn; }
    }
    tgi = bn; ov = fmaxf(best, 0.0f);
  } else if (cnt == 1) {
    tgi = __ffsll((unsigned long long)selbits[i]) - 1;
    float4 g = ((const float4*)gt_bboxes)[b * NGT + tgi];
    ov = fmaxf(ciou_gt_pd(g.x, g.y, g.z, g.w, pb.x, pb.y, pb.z, pb.w), 0.0f);
  }

  int r = b * NGT + tgi;
  int lbl = gt_labels[r]; if (lbl < 0) lbl = 0;
  float alignv = 0.0f;
  if (fg) {
    float sc = pd_scores[((size_t)b * NA + a) * NC + lbl];
    float o2 = ov * ov;
    alignv = sc * o2 * o2 * o2;
    // values >= 0: int-ordered atomicMax == float max
    atomicMax((int*)(pos_align + r), __float_as_int(alignv));
    atomicMax((int*)(pos_ov + r), __float_as_int(ov));
  }
  ws_align[i] = alignv;
  ws_tgi[i] = tgi;
  ws_fg[i] = fg;

  outL[i] = (float)lbl;
  ((float4*)outB)[i] = ((const float4*)gt_bboxes)[r];
  outF[i] = fg ? 1.0f : 0.0f;
  outG[i] = (float)tgi;
}

// ---------------------------------------------------------------- Phase B2 (WMMA): target_scores = (mask_pos . norm)^T x one_hot(labels)
// One wave owns a 16-anchor tile.  S(16x80) = M(16x64) x T(64x80), K-steps of 4 via
// V_WMMA_F32_16X16X4_F32.  M has exactly one nonzero per row (norm at tgi) so the
// f32 accumulation is exact.  VGPR layouts per CDNA5 ISA 7.12.2:
//   A 16x4 : lane&15 = M row; lane half selects K {0,1} vs {2,3} across the 2 VGPRs
//   B 4x16 : lane&15 = N col; lane half selects K half (mirrors A)
//   D 16x16: VGPR j -> M=j (lanes 0-15) / M=j+8 (lanes 16-31), N = lane&15
__global__ __launch_bounds__(256) void phaseB2W(const int* __restrict__ gt_labels,
                                                const float* __restrict__ pos_align,
                                                const float* __restrict__ pos_ov,
                                                const float* __restrict__ ws_align,
                                                const int*   __restrict__ ws_tgi,
                                                const int*   __restrict__ ws_fg,
                                                float* __restrict__ outS) {
  const int lane = threadIdx.x & 31;
  const int wave = threadIdx.x >> 5;
  const int i0 = blockIdx.x * 128 + wave * 16;   // 16-anchor tile base (NA % 16 == 0: no batch straddle)
  const int b = i0 / NA;
  const int r0 = b * NGT;
  const int m = lane & 15;
  const int kh = (lane >> 4) * 2;                // K offset contributed by lane half

  // this lane's A-row data (lanes 16-31 duplicate the loads of lanes 0-15; L1 hit)
  const int im = i0 + m;
  const int tgi = ws_tgi[im];
  float norm = 0.0f;
  if (ws_fg[im]) {
    int r = r0 + tgi;
    norm = ws_align[im] * pos_ov[r] / (pos_align[r] + 1e-9f);
  }

  for (int ct = 0; ct < NC; ct += 16) {
    const int n = ct + m;                        // this lane's B column (class id)
    v8f acc = {0.0f, 0.0f, 0.0f, 0.0f, 0.0f, 0.0f, 0.0f, 0.0f};
#pragma unroll
    for (int k0 = 0; k0 < NGT; k0 += 4) {
      const int ka = k0 + kh;
      const int kb = ka + 1;
      v2f A, B;
      A.x = (tgi == ka) ? norm : 0.0f;
      A.y = (tgi == kb) ? norm : 0.0f;
      int l0 = gt_labels[r0 + ka]; if (l0 < 0) l0 = 0;
      int l1 = gt_labels[r0 + kb]; if (l1 < 0) l1 = 0;
      B.x = (l0 == n) ? 1.0f : 0.0f;
      B.y = (l1 == n) ? 1.0f : 0.0f;
      acc = __builtin_amdgcn_wmma_f32_16x16x4_f32(false, A, false, B, (short)0, acc,
                                                  false, false);
    }
    // write D: lanes 0-15 rows i0+j, lanes 16-31 rows i0+8+j; 16 lanes x 4B contiguous
    float* base = outS + (size_t)(i0 + (lane >> 4) * 8) * NC + n;
#pragma unroll
    for (int j = 0; j < 8; ++j) {
      base[(size_t)j * NC] = acc[j];
    }
  }
}

// ---------------------------------------------------------------- launcher
extern "C" void kernel_launch(void* const* d_in, const int* in_sizes, int n_in,
                              void* d_out, int out_size, void* d_ws, size_t ws_size,
                              hipStream_t stream) {
  const float* pd_scores = (const float*)d_in[0];
  const float* pd_bboxes = (const float*)d_in[1];
  const float* anc       = (const float*)d_in[2];
  const int*   gt_labels = (const int*)d_in[3];
  const float* gt_bboxes = (const float*)d_in[4];
  const float* mask_gt   = (const float*)d_in[5];

  float* out = (float*)d_out;
  float* outL = out;                                  // target_labels  (bs,na)
  float* outB = out + BS * NA;                        // target_bboxes  (bs,na,4)
  float* outS = out + BS * NA * 5;                    // target_scores  (bs,na,80)
  float* outF = out + BS * NA * 85;                   // fg_mask        (bs,na)
  float* outG = out + BS * NA * 86;                   // target_gt_idx  (bs,na)

  char* w = (char*)d_ws;
  u64* selbits    = (u64*)w;   w += (size_t)BS * NA * 8;
  int* selcount   = (int*)w;   w += (size_t)BS * NA * 4;
  float* pos_align = (float*)w; w += (size_t)BS * NGT * 4;
  float* pos_ov    = (float*)w; w += (size_t)BS * NGT * 4;
  float* ws_align  = (float*)w; w += (size_t)BS * NA * 4;
  int* ws_tgi      = (int*)w;   w += (size_t)BS * NA * 4;
  int* ws_fg       = (int*)w;   w += (size_t)BS * NA * 4;

  const int zero_words = (BS * NA * 8 + BS * NA * 4 + 2 * BS * NGT * 4) / 4;  // 405248
  initWS<<<(zero_words + 255) / 256, 256, 0, stream>>>((u32*)d_ws, zero_words);

  phaseA<<<BS * NGT, ABLK, 0, stream>>>(pd_scores, pd_bboxes, anc, gt_labels, gt_bboxes,
                                        mask_gt, selcount, selbits);

  const int nanch = BS * NA;  // 134400 = 525 * 256 = 1050 * 128
  phaseB1<<<nanch / 256, 256, 0, stream>>>(pd_scores, pd_bboxes, anc, gt_labels, gt_bboxes,
                                           mask_gt, selcount, selbits, pos_align, pos_ov,
                                           ws_align, ws_tgi, ws_fg, outL, outB, outF, outG);

  phaseB2W<<<nanch / 128, 256, 0, stream>>>(gt_labels, pos_align, pos_ov,
                                            ws_align, ws_tgi, ws_fg, outS);
}